// T3A_83554293776646
// MI455X (gfx1250) — compile-verified
//
#include <hip/hip_runtime.h>
#include <hip/hip_bf16.h>
#include <math.h>

typedef __attribute__((ext_vector_type(2))) float v2f;
typedef __attribute__((ext_vector_type(8))) float v8f;

#define KMAX 128

// -------------------------------------------------------------------------
// Generic fp32 WMMA GEMM:  Out[r, c] = sum_k A[r, k] * Bm[c, k]
// A is the concatenation of A0 (A0_rows rows) and A1 (rows A0_rows..A_rows).
// Each wave computes a 32x32 tile via 4 v_wmma_f32_16x16x4_f32 chains.
// Row/col edges: loads clamp the row index (EXEC stays all-ones for WMMA),
// stores are guarded by out_rows / out_cols.
// -------------------------------------------------------------------------
__global__ __launch_bounds__(128)
void gemm_wmma_f32(const float* __restrict__ A0, int A0_rows,
                   const float* __restrict__ A1, int A_rows,
                   const float* __restrict__ Bm, int B_rows,
                   float* __restrict__ Out, int ldOut,
                   int out_cols, int out_rows, int D)
{
    const int wave = threadIdx.x >> 5;
    const int lane = threadIdx.x & 31;
    const int row0 = (blockIdx.y * 4 + wave) * 32;
    const int col0 = blockIdx.x * 32;
    if (row0 >= out_rows) return;   // wave-uniform exit

    const int lr   = lane & 15;
    const int koff = (lane >> 4) << 1;   // 0 for lanes 0-15, 2 for lanes 16-31

    int ar0 = row0 + lr;       if (ar0 >= A_rows) ar0 = A_rows - 1;
    int ar1 = row0 + 16 + lr;  if (ar1 >= A_rows) ar1 = A_rows - 1;
    const float* ap0 = (ar0 < A0_rows) ? (A0 + (size_t)ar0 * D)
                                       : (A1 + (size_t)(ar0 - A0_rows) * D);
    const float* ap1 = (ar1 < A0_rows) ? (A0 + (size_t)ar1 * D)
                                       : (A1 + (size_t)(ar1 - A0_rows) * D);

    int br0 = col0 + lr;       if (br0 >= B_rows) br0 = B_rows - 1;
    int br1 = col0 + 16 + lr;  if (br1 >= B_rows) br1 = B_rows - 1;
    const float* bp0 = Bm + (size_t)br0 * D;
    const float* bp1 = Bm + (size_t)br1 * D;

    v8f acc00 = {}, acc01 = {}, acc10 = {}, acc11 = {};

#pragma unroll 4
    for (int k = 0; k < D; k += 4) {
        v2f a0 = *(const v2f*)(ap0 + k + koff);
        v2f a1 = *(const v2f*)(ap1 + k + koff);
        v2f b0 = *(const v2f*)(bp0 + k + koff);
        v2f b1 = *(const v2f*)(bp1 + k + koff);
        acc00 = __builtin_amdgcn_wmma_f32_16x16x4_f32(false, a0, false, b0,
                                                      (short)0, acc00, false, false);
        acc01 = __builtin_amdgcn_wmma_f32_16x16x4_f32(false, a0, false, b1,
                                                      (short)0, acc01, false, false);
        acc10 = __builtin_amdgcn_wmma_f32_16x16x4_f32(false, a1, false, b0,
                                                      (short)0, acc10, false, false);
        acc11 = __builtin_amdgcn_wmma_f32_16x16x4_f32(false, a1, false, b1,
                                                      (short)0, acc11, false, false);
    }

    // C/D layout: VGPR j -> row M = j + (lane>=16 ? 8 : 0), col N = lane & 15
    const int rofs = (lane >> 4) << 3;
    auto store_tile = [&](const v8f& c, int rbase, int cbase) {
        int col = cbase + lr;
        if (col >= out_cols) return;
#pragma unroll
        for (int j = 0; j < 8; ++j) {
            int r = rbase + rofs + j;
            if (r < out_rows) Out[(size_t)r * ldOut + col] = c[j];
        }
    };
    store_tile(acc00, row0,      col0);
    store_tile(acc01, row0,      col0 + 16);
    store_tile(acc10, row0 + 16, col0);
    store_tile(acc11, row0 + 16, col0 + 16);
}

// -------------------------------------------------------------------------
// One wave per row: entropy of softmax + argmax (ties -> smallest index)
//   ent = M + log(S) - T/S  with  S = sum e^{x-M}, T = sum x e^{x-M}
// -------------------------------------------------------------------------
__global__ __launch_bounds__(256)
void entropy_argmax_kernel(const float* __restrict__ P, int ldP, int C,
                           int N, float* __restrict__ ent, int* __restrict__ yhat)
{
    const int wave = threadIdx.x >> 5;
    const int lane = threadIdx.x & 31;
    const int n = blockIdx.x * 8 + wave;
    if (n >= N) return;
    const float* row = P + (size_t)n * ldP;

    float m = -3.0e38f; int mi = 0x7fffffff;
    for (int c = lane; c < C; c += 32) {
        float v = row[c];
        if (v > m) { m = v; mi = c; }
    }
    for (int off = 16; off; off >>= 1) {
        float vm = __shfl_xor(m, off, 32);
        int   vi = __shfl_xor(mi, off, 32);
        if (vm > m || (vm == m && vi < mi)) { m = vm; mi = vi; }
    }

    float S = 0.f, T = 0.f;
    for (int c = lane; c < C; c += 32) {
        float x = row[c];
        float e = __expf(x - m);
        S += e; T += e * x;
    }
    for (int off = 16; off; off >>= 1) {
        S += __shfl_xor(S, off, 32);
        T += __shfl_xor(T, off, 32);
    }
    if (lane == 0) {
        ent[n]  = m + __logf(S) - T / S;
        yhat[n] = mi;
    }
}

// -------------------------------------------------------------------------
// One wave per support row: rinv[n] = 1 / max(||s_n||, 1e-12)
// -------------------------------------------------------------------------
__global__ __launch_bounds__(256)
void row_rnorm_kernel(const float* __restrict__ Wm, int C,
                      const float* __restrict__ Zm,
                      int N, int D, float* __restrict__ rinv)
{
    const int wave = threadIdx.x >> 5;
    const int lane = threadIdx.x & 31;
    const int n = blockIdx.x * 8 + wave;
    if (n >= N) return;
    const float* row = (n < C) ? (Wm + (size_t)n * D) : (Zm + (size_t)(n - C) * D);
    float s = 0.f;
    for (int d = lane; d < D; d += 32) { float v = row[d]; s += v * v; }
    for (int off = 16; off; off >>= 1) s += __shfl_xor(s, off, 32);
    if (lane == 0) rinv[n] = 1.0f / fmaxf(sqrtf(s), 1e-12f);
}

// -------------------------------------------------------------------------
// One block per class c: deterministic rank-based top-K selection of the
// K lowest-entropy supports with yhat==c, accumulate normalized rows into
// LDS, normalize the column, write WT[c, :].  Rows c >= C are zero pad.
// -------------------------------------------------------------------------
__global__ __launch_bounds__(256)
void select_accumulate_kernel(const float* __restrict__ Wm, int C,
                              const float* __restrict__ Zm,
                              int N, int D,
                              const float* __restrict__ ent,
                              const int* __restrict__ yhat,
                              const float* __restrict__ rinv,
                              const int* __restrict__ Kptr,
                              float* __restrict__ WT, int ldWT)
{
    __shared__ float acc[2048];
    __shared__ int   sel[KMAX];
    __shared__ float red[8];

    const int c = blockIdx.x;
    int K = *Kptr; if (K > KMAX) K = KMAX; if (K < 0) K = 0;

    for (int d = threadIdx.x; d < D; d += 256) acc[d] = 0.f;
    for (int s = threadIdx.x; s < KMAX; s += 256) sel[s] = -1;
    __syncthreads();

    if (c < C) {
        // rank of member n among class-c members, ordered by (ent, idx)
        for (int n = threadIdx.x; n < N; n += 256) {
            if (yhat[n] != c) continue;
            float e = ent[n];
            int rank = 0;
            for (int m = 0; m < N; ++m) {
                if (yhat[m] != c) continue;
                float em = ent[m];
                if (em < e || (em == e && m < n)) {
                    if (++rank >= K) break;
                }
            }
            if (rank < K) sel[rank] = n;   // ranks are unique -> race-free
        }
        __syncthreads();

        for (int s = 0; s < K; ++s) {
            int n = sel[s];
            if (n < 0) break;              // uniform across block
            float r = rinv[n];
            const float* row = (n < C) ? (Wm + (size_t)n * D)
                                       : (Zm + (size_t)(n - C) * D);
            for (int d = threadIdx.x; d < D; d += 256)
                acc[d] += row[d] * r;      // thread-private d stripes: no race
        }
    }
    __syncthreads();

    // column L2 norm
    float s = 0.f;
    for (int d = threadIdx.x; d < D; d += 256) { float v = acc[d]; s += v * v; }
    for (int off = 16; off; off >>= 1) s += __shfl_xor(s, off, 32);
    const int wave = threadIdx.x >> 5;
    const int lane = threadIdx.x & 31;
    if (lane == 0) red[wave] = s;
    __syncthreads();
    float tot = 0.f;
#pragma unroll
    for (int i = 0; i < 8; ++i) tot += red[i];
    float ci = 1.0f / fmaxf(sqrtf(tot), 1e-12f);

    for (int d = threadIdx.x; d < D; d += 256)
        WT[(size_t)c * ldWT + d] = acc[d] * ci;
}

// -------------------------------------------------------------------------
extern "C" void kernel_launch(void* const* d_in, const int* in_sizes, int n_in,
                              void* d_out, int out_size, void* d_ws, size_t ws_size,
                              hipStream_t stream)
{
    const float* z  = (const float*)d_in[0];   // [B, D]
    const float* W  = (const float*)d_in[1];   // [C, D]
    const int*   fK = (const int*)d_in[2];     // filter_K (device scalar)

    const int Bn = 4096, Dd = 2048, Cc = 1000;
    const int N     = Cc + Bn;            // 5096
    const int rowsP = (N + 15) & ~15;     // 5104
    const int ldP   = (Cc + 15) & ~15;    // 1008
    const int WTrows = ldP;               // 1008 (rows >= Cc are zero)

    float* P    = (float*)d_ws;                       // rowsP * ldP
    float* WT   = P  + (size_t)rowsP * ldP;           // WTrows * Dd
    float* ent  = WT + (size_t)WTrows * Dd;           // rowsP
    int*   yhat = (int*)(ent + rowsP);                // rowsP
    float* rinv = (float*)(yhat + rowsP);             // rowsP

    // 1) P = concat(W, z) @ W^T   (padded cols hold garbage, never read)
    {
        dim3 blk(128);
        dim3 grd((ldP + 31) / 32, (((rowsP + 31) / 32) + 3) / 4);
        gemm_wmma_f32<<<grd, blk, 0, stream>>>(W, Cc, z, N, W, Cc,
                                               P, ldP, ldP, rowsP, Dd);
    }
    // 2) entropy + argmax per row (scans only c < Cc)
    entropy_argmax_kernel<<<dim3((N + 7) / 8), dim3(256), 0, stream>>>(
        P, ldP, Cc, N, ent, yhat);
    // 3) support row norms
    row_rnorm_kernel<<<dim3((N + 7) / 8), dim3(256), 0, stream>>>(
        W, Cc, z, N, Dd, rinv);
    // 4) per-class top-K selection + normalized accumulation -> WT [WTrows, D]
    select_accumulate_kernel<<<dim3(WTrows), dim3(256), 0, stream>>>(
        W, Cc, z, N, Dd, ent, yhat, rinv, fK, WT, Dd);
    // 5) out = z @ WT^T  -> d_out [B, Cc]
    {
        dim3 blk(128);
        dim3 grd((ldP + 31) / 32, (((Bn + 31) / 32) + 3) / 4);
        gemm_wmma_f32<<<grd, blk, 0, stream>>>(z, Bn, z, Bn, WT, WTrows,
                                               (float*)d_out, Cc, Cc, Bn, Dd);
    }
}